// _PartitionLayer_16750372455195
// MI455X (gfx1250) — compile-verified
//
#include <hip/hip_runtime.h>

typedef float v2f __attribute__((ext_vector_type(2)));
typedef float v4f __attribute__((ext_vector_type(4)));
typedef float v8f __attribute__((ext_vector_type(8)));

#define BATCH 131072
#define DFEAT 128
#define RCOLS 32
#define ROWS_PER_BLOCK 128
#define THREADS 256

// 8 waves per block; each wave computes a 16-row tile of x1 (16x32) and x2 (16x32)
// with V_WMMA_F32_16X16X4_F32, stages them in LDS, then the whole block streams
// the 128x1024 outer-sum tile to HBM with coalesced non-temporal b128 stores.
__global__ __launch_bounds__(THREADS) void partition_layer_kernel(
    const float* __restrict__ data,
    const float* __restrict__ W1,
    const float* __restrict__ W2,
    float* __restrict__ out)
{
    __shared__ alignas(16) float x1s[ROWS_PER_BLOCK * RCOLS];
    __shared__ alignas(16) float x2s[ROWS_PER_BLOCK * RCOLS];

    const int tid  = threadIdx.x;
    const int wave = tid >> 5;     // 0..7
    const int lane = tid & 31;     // 0..31
    const int lo   = lane & 15;    // 0..15  (A: M index, B: N index)
    const int hi   = lane >> 4;    // 0/1    (selects K pair within a 4-wide K step)

    const int blockRow0 = blockIdx.x * ROWS_PER_BLOCK;
    const int waveRow0  = blockRow0 + wave * 16;

    v8f acc00 = {}; // x1 cols  0..15
    v8f acc01 = {}; // x1 cols 16..31
    v8f acc10 = {}; // x2 cols  0..15
    v8f acc11 = {}; // x2 cols 16..31

    // A fragment base: lane holds data[M = waveRow0+lo][k0 + 2*hi + {0,1}]
    const float* arow = data + (size_t)(waveRow0 + lo) * DFEAT + hi * 2;

    for (int k0 = 0; k0 < DFEAT; k0 += 4) {
        v2f a;
        a.x = arow[k0 + 0];
        a.y = arow[k0 + 1];

        const int krow = k0 + hi * 2;
        // B fragment: lane holds W[K = krow + {0,1}][N = lo (+16)]
        v2f b00, b01, b10, b11;
        b00.x = W1[(size_t)(krow    ) * RCOLS + lo];
        b00.y = W1[(size_t)(krow + 1) * RCOLS + lo];
        b01.x = W1[(size_t)(krow    ) * RCOLS + 16 + lo];
        b01.y = W1[(size_t)(krow + 1) * RCOLS + 16 + lo];
        b10.x = W2[(size_t)(krow    ) * RCOLS + lo];
        b10.y = W2[(size_t)(krow + 1) * RCOLS + lo];
        b11.x = W2[(size_t)(krow    ) * RCOLS + 16 + lo];
        b11.y = W2[(size_t)(krow + 1) * RCOLS + 16 + lo];

        // 8-arg form: (neg_a, A, neg_b, B, c_mod, C, reuse_a, reuse_b)
        acc00 = __builtin_amdgcn_wmma_f32_16x16x4_f32(false, a, false, b00, (short)0, acc00, false, false);
        acc01 = __builtin_amdgcn_wmma_f32_16x16x4_f32(false, a, false, b01, (short)0, acc01, false, false);
        acc10 = __builtin_amdgcn_wmma_f32_16x16x4_f32(false, a, false, b10, (short)0, acc10, false, false);
        acc11 = __builtin_amdgcn_wmma_f32_16x16x4_f32(false, a, false, b11, (short)0, acc11, false, false);
    }

    // C/D layout: VGPR v, lane l -> element [M = v + 8*hi][N = lo] of the 16x16 tile.
    {
        const int rbase = wave * 16 + 8 * hi;
        #pragma unroll
        for (int v = 0; v < 8; ++v) {
            x1s[(rbase + v) * RCOLS + lo]      = acc00[v];
            x1s[(rbase + v) * RCOLS + 16 + lo] = acc01[v];
            x2s[(rbase + v) * RCOLS + lo]      = acc10[v];
            x2s[(rbase + v) * RCOLS + 16 + lo] = acc11[v];
        }
    }
    __syncthreads();

    // Outer sum + streaming store. Thread t owns out[row][4t .. 4t+3]:
    //   i  = (4t)>>5 = t>>3  (x1 column), j0 = (4t)&31 = (t&7)*4 (x2 column base)
    const int i  = tid >> 3;
    const int j0 = (tid & 7) * 4;
    float* outp = out + (size_t)blockRow0 * (RCOLS * RCOLS) + tid * 4;

    for (int r = 0; r < ROWS_PER_BLOCK; ++r) {
        const float a1 = x1s[r * RCOLS + i];
        const v4f  b2 = *(const v4f*)&x2s[r * RCOLS + j0];
        v4f o;
        o.x = a1 + b2.x;
        o.y = a1 + b2.y;
        o.z = a1 + b2.z;
        o.w = a1 + b2.w;
        __builtin_nontemporal_store(o, (v4f*)(outp + (size_t)r * (RCOLS * RCOLS)));
    }
}

extern "C" void kernel_launch(void* const* d_in, const int* in_sizes, int n_in,
                              void* d_out, int out_size, void* d_ws, size_t ws_size,
                              hipStream_t stream) {
    const float* data = (const float*)d_in[0];
    const float* W1   = (const float*)d_in[1];
    const float* W2   = (const float*)d_in[2];
    float* out        = (float*)d_out;

    dim3 grid(BATCH / ROWS_PER_BLOCK);   // 1024 blocks
    dim3 block(THREADS);                 // 8 waves (wave32)
    partition_layer_kernel<<<grid, block, 0, stream>>>(data, W1, W2, out);
}